// FullOnlineHadamardHook_46153718563372
// MI455X (gfx1250) — compile-verified
//
#include <hip/hip_runtime.h>
#include <math.h>
#include <stdint.h>

typedef __attribute__((ext_vector_type(2))) float v2f;
typedef __attribute__((ext_vector_type(8))) float v8f;

#define N_TOT    14336      // row length
#define KROW     28         // had_K dimension
#define M_LEN    512        // FWHT length (N_TOT / KROW)
#define LDSTRIDE 516        // 512 + 4 floats pad -> de-alias LDS banks across rows
#define NTHREADS 512        // 16 waves (wave32)

__global__ __launch_bounds__(NTHREADS)
void fht_hadk_kernel(const float* __restrict__ x,
                     const float* __restrict__ hadK,
                     float* __restrict__ out)
{
    __shared__ __align__(16) float lds[KROW * LDSTRIDE];   // 57,792 bytes

    const int tid  = threadIdx.x;
    const int lane = tid & 31;
    const int wv   = tid >> 5;               // wave id 0..15
    const int lh   = lane & 15;              // lane within half-wave
    const bool hi  = (lane >= 16);

    const long row_base = (long)blockIdx.x * N_TOT;

    // ---------------- Phase 1: async global -> LDS (bypass VGPRs) -------------
    // GLOBAL_LOAD_ASYNC_TO_LDS_B128: per lane, copies 16 bytes from the global address in vaddr to the LDS address in vdst.
    // LDS byte offset = low 32 bits of the flat shared-pointer address.
    {
        const unsigned lds_base =
            (unsigned)(unsigned long long)(uintptr_t)(const void*)lds;
        #pragma unroll
        for (int it = 0; it < 7; ++it) {
            int v4 = tid + it * NTHREADS;            // 0..3583 float4 chunks
            int e  = v4 << 2;                        // element index
            int r  = e >> 9;                         // row 0..27
            int c  = e & 511;                        // col 0..511
            unsigned loff = lds_base + (unsigned)((r * LDSTRIDE + c) * 4);
            unsigned long long gaddr =
                (unsigned long long)(uintptr_t)(x + row_base + (long)e);
            asm volatile("global_load_async_to_lds_b128 %0, %1, off"
                         :: "v"(loff), "v"(gaddr) : "memory");
        }
        asm volatile("s_wait_asynccnt 0x0" ::: "memory");
    }
    __syncthreads();

    // ---------------- Phase 2: FWHT-512 on each of 28 rows --------------------
    // 28 rows * 256 pairs = 7168 butterflies per stage -> 14 per thread.
    for (int h = 1; h < M_LEN; h <<= 1) {
        #pragma unroll
        for (int r = 0; r < 14; ++r) {
            int p  = tid + r * NTHREADS;     // 0..7167, all pairs disjoint
            int rw = p >> 8;                 // row 0..27
            int q  = p & 255;                // pair index within row
            int i  = ((q & ~(h - 1)) << 1) | (q & (h - 1));
            int a0 = rw * LDSTRIDE + i;
            float a = lds[a0];
            float b = lds[a0 + h];
            lds[a0]     = a + b;
            lds[a0 + h] = a - b;
        }
        __syncthreads();
    }

    // ---------------- Phase 3: load A fragments (H * 1/sqrt(N)) ---------------
    // V_WMMA_F32_16X16X4_F32 A layout (16x4 fp32, 2 VGPRs):
    //   lanes 0-15 : VGPR0 = A[M=lane][K=0], VGPR1 = A[M=lane][K=1]
    //   lanes 16-31: VGPR0 = A[M=lane-16][K=2], VGPR1 = A[M=lane-16][K=3]
    const float scale = 8.3519792e-3f;       // 1/sqrt(14336)
    v2f afrag[2][7];
    #pragma unroll
    for (int mi = 0; mi < 2; ++mi) {
        int m = mi * 16 + lh;                // global M row of H (pad >= 28 with 0)
        #pragma unroll
        for (int ks = 0; ks < 7; ++ks) {
            int k0 = ks * 4 + (hi ? 2 : 0);  // k0, k0+1 <= 27 always
            float a0 = 0.0f, a1 = 0.0f;
            if (m < KROW) {
                a0 = hadK[m * KROW + k0]     * scale;
                a1 = hadK[m * KROW + k0 + 1] * scale;
            }
            afrag[mi][ks].x = a0;
            afrag[mi][ks].y = a1;
        }
    }

    // ---------------- Phase 4: OUT = H * X  via fp32 WMMA ---------------------
    // 2 M-tiles x 32 N-tiles x 7 K-slices. Each wave owns N-tiles {wv, wv+16}
    // and both M-tiles, so every B fragment feeds two WMMAs.
    v8f acc[2][2];                           // [ntIdx][mi]
    #pragma unroll
    for (int a = 0; a < 2; ++a)
        #pragma unroll
        for (int b = 0; b < 2; ++b)
            acc[a][b] = (v8f){0.f,0.f,0.f,0.f,0.f,0.f,0.f,0.f};

    #pragma unroll
    for (int nti = 0; nti < 2; ++nti) {
        int nt   = wv + nti * 16;            // N-tile 0..31
        int ncol = nt * 16 + lh;             // column this lane reads
        #pragma unroll
        for (int ks = 0; ks < 7; ++ks) {
            // B layout (4x16 fp32, 2 VGPRs), mirroring the A split:
            //   lanes 0-15 : VGPR0 = B[K=0][N=lane], VGPR1 = B[K=1][N=lane]
            //   lanes 16-31: VGPR0 = B[K=2][...],    VGPR1 = B[K=3][...]
            int k0 = ks * 4 + (hi ? 2 : 0);
            v2f bfrag;
            bfrag.x = lds[ k0      * LDSTRIDE + ncol];
            bfrag.y = lds[(k0 + 1) * LDSTRIDE + ncol];
            acc[nti][0] = __builtin_amdgcn_wmma_f32_16x16x4_f32(
                false, afrag[0][ks], false, bfrag, (short)0, acc[nti][0], false, false);
            acc[nti][1] = __builtin_amdgcn_wmma_f32_16x16x4_f32(
                false, afrag[1][ks], false, bfrag, (short)0, acc[nti][1], false, false);
        }
    }

    // ---------------- Phase 5: accumulators -> global -------------------------
    // C/D layout: VGPR v, lanes 0-15 -> M = v, lanes 16-31 -> M = v + 8.
    #pragma unroll
    for (int nti = 0; nti < 2; ++nti) {
        int n = (wv + nti * 16) * 16 + lh;
        #pragma unroll
        for (int mi = 0; mi < 2; ++mi) {
            #pragma unroll
            for (int v = 0; v < 8; ++v) {
                int M = mi * 16 + v + (hi ? 8 : 0);
                if (M < KROW) {
                    out[row_base + (long)M * M_LEN + n] = acc[nti][mi][v];
                }
            }
        }
    }
}

extern "C" void kernel_launch(void* const* d_in, const int* in_sizes, int n_in,
                              void* d_out, int out_size, void* d_ws, size_t ws_size,
                              hipStream_t stream)
{
    const float* x    = (const float*)d_in[0];   // (B, S, N) fp32
    const float* hadK = (const float*)d_in[1];   // (28, 28) fp32
    float* out        = (float*)d_out;

    int rows = in_sizes[0] / N_TOT;              // B*S = 8192
    hipLaunchKernelGGL(fht_hadk_kernel, dim3(rows), dim3(NTHREADS), 0, stream,
                       x, hadK, out);
}